// LIFLayer_17274358464685
// MI455X (gfx1250) — compile-verified
//
#include <hip/hip_runtime.h>

// LIF recurrence, T=2048 steps x N=32768 neurons.
// Memory-bound: 256MB read + 512MB write => ~33us floor at 23.3 TB/s.
// CDNA5 path: deep async global->LDS prefetch (ASYNCcnt-tracked ring, 48 steps
// deep => 128 blocks * 48KB ~= 6MB of reads in flight), software-pipelined
// LDS consumption (ds_load issued one step ahead), NT hints on both streams.

#define T_STEPS 2048
#define NEURONS 32768
#define BLOCK   256
#define STAGES  48                      // ring slots; outstanding ASYNCcnt <= 48 (< 63 cap)
#define ALPHA   0.9512294245007140f     // exp(-1/20)
#define BETA    (1.0f - ALPHA)

__global__ __launch_bounds__(BLOCK) void lif_async_kernel(
    const float* __restrict__ currents,   // [T, N]
    const float* __restrict__ v0,         // [N]
    float* __restrict__ spikes,           // [T, N]
    float* __restrict__ volts)            // [T, N]
{
    __shared__ float ring[STAGES * BLOCK];   // 48 KB LDS ring, slot = one time step

    const int tid = threadIdx.x;
    const int gn  = blockIdx.x * BLOCK + tid;     // neuron owned by this thread

    // 32-bit LDS byte address of this lane's cell in slot 0.
    unsigned lds0 = (unsigned)(unsigned long long)&ring[tid];

    // Per-lane global source address, advanced one row (N floats) per issue.
    const float* gsrc = currents + gn;

    // ---- Prologue: fill the ring. ASYNCcnt ramps to STAGES. ----
    #pragma unroll
    for (int s = 0; s < STAGES; ++s) {
        unsigned lslot = lds0 + (unsigned)(s * BLOCK * 4);
        asm volatile("global_load_async_to_lds_b32 %0, %1, off th:TH_LOAD_NT"
                     :: "v"(lslot), "v"(gsrc)
                     : "memory");
        gsrc += NEURONS;
    }

    float v = v0[gn];
    float* sp = spikes + gn;
    float* vp = volts  + gn;

    // ---- Startup: step 0 has landed; prime the register pipeline. ----
    asm volatile("s_wait_asynccnt 47" ::: "memory");
    float c = ring[tid];                 // current for step 0

    int slot = 0;
    // ---- Main loop: 2000 iterations, refill + one-step-ahead LDS read. ----
    #pragma unroll 4
    for (int t = 0; t < T_STEPS - STAGES; ++t) {
        // Retire all pending LDS reads (c now in a register) before the async
        // refill below may overwrite this slot. Latency of that ds_load was
        // hidden behind the previous iteration's math + stores.
        asm volatile("s_wait_dscnt 0" ::: "memory");

        // Refill the slot being consumed with step t+STAGES.
        unsigned lslot = lds0 + (unsigned)(slot * (BLOCK * 4));
        asm volatile("global_load_async_to_lds_b32 %0, %1, off th:TH_LOAD_NT"
                     :: "v"(lslot), "v"(gsrc)
                     : "memory");
        gsrc += NEURONS;

        // With the refill issued, issued = t+1+STAGES, so outstanding <= 47
        // implies the load for step t+1 has landed. Kick off its LDS read now;
        // we consume it next iteration.
        int nslot = (slot + 1 == STAGES) ? 0 : slot + 1;
        asm volatile("s_wait_asynccnt 47" ::: "memory");
        float cn = ring[nslot * BLOCK + tid];   // ds_load_b32, waited next iter

        // leaky integrate -> threshold -> reset for step t
        v = ALPHA * v + BETA * c;
        float s = (v >= 1.0f) ? 1.0f : 0.0f;
        v       = (v >= 1.0f) ? 0.0f : v;

        __builtin_nontemporal_store(s, sp);
        __builtin_nontemporal_store(v, vp);
        sp += NEURONS;
        vp += NEURONS;

        c = cn;
        slot = nslot;
    }

    // ---- Drain: last 48 steps are all resident in LDS already. ----
    asm volatile("s_wait_asynccnt 0" ::: "memory");
    #pragma unroll 4
    for (int t = T_STEPS - STAGES; t < T_STEPS; ++t) {
        v = ALPHA * v + BETA * c;
        float s = (v >= 1.0f) ? 1.0f : 0.0f;
        v       = (v >= 1.0f) ? 0.0f : v;

        __builtin_nontemporal_store(s, sp);
        __builtin_nontemporal_store(v, vp);
        sp += NEURONS;
        vp += NEURONS;

        int nslot = (slot + 1 == STAGES) ? 0 : slot + 1;
        if (t + 1 < T_STEPS)
            c = ring[nslot * BLOCK + tid];
        slot = nslot;
    }
}

extern "C" void kernel_launch(void* const* d_in, const int* in_sizes, int n_in,
                              void* d_out, int out_size, void* d_ws, size_t ws_size,
                              hipStream_t stream) {
    const float* currents = (const float*)d_in[0];   // [T*N] f32
    const float* v0       = (const float*)d_in[1];   // [N]   f32

    float* spikes = (float*)d_out;                               // first output
    float* volts  = (float*)d_out + (size_t)T_STEPS * NEURONS;   // second output

    dim3 grid(NEURONS / BLOCK);   // 128 blocks
    dim3 block(BLOCK);            // 256 threads = 8 wave32 waves
    hipLaunchKernelGGL(lif_async_kernel, grid, block, 0, stream,
                       currents, v0, spikes, volts);
}